// BimodalAttention_50002009260177
// MI455X (gfx1250) — compile-verified
//
#include <hip/hip_runtime.h>

// Problem constants (fixed by the reference harness).
#define B_DIM 128
#define L_DIM 512
#define D_DIM 256
#define T_DIM 64
#define G_DIM 8   // L / T

typedef __attribute__((ext_vector_type(2))) float v2f;
typedef __attribute__((ext_vector_type(4))) float f4;
typedef __attribute__((ext_vector_type(8))) float v8f;

__device__ __forceinline__ float sigmoidf(float x) {
    return 1.0f / (1.0f + __expf(-x));
}

// ---------------------------------------------------------------------------
// Pass A: per (b, chunk-of-64-rows) block, 64 threads, each owning 4 columns
// (b128 loads). Produces:
//   hm[b][t]          = sum_{l in group t, all d} (a+v) / 4096  (mean_D of hw)
//   wm_part[b][ch][d] = sum over the chunk's 64 rows of (a+v)
// Default (RT) loads deliberately warm L2 (192 MB) so Pass C's re-read of the
// inputs hits cache.
// ---------------------------------------------------------------------------
__global__ void pool_reduce_kernel(const float* __restrict__ aco,
                                   const float* __restrict__ vis,
                                   float* __restrict__ hm,
                                   float* __restrict__ wm_part) {
    const int b   = blockIdx.x >> 3;   // 128 b
    const int ch  = blockIdx.x & 7;    // 8 chunks of 64 rows
    const int tid = threadIdx.x;       // 64 threads
    const int d4  = tid << 2;          // 4 columns each
    __shared__ float red[64];

    f4 colsum = {0.f, 0.f, 0.f, 0.f};
    const size_t rowbase = (size_t)(b * L_DIM + ch * 64) * D_DIM + d4;

    #pragma unroll 1
    for (int g = 0; g < 8; ++g) {      // 8 pooling groups per chunk
        f4 gsum4 = {0.f, 0.f, 0.f, 0.f};
        #pragma unroll
        for (int i = 0; i < 8; ++i) {
            const size_t off = rowbase + (size_t)(g * 8 + i) * D_DIM;
            const f4 av = *reinterpret_cast<const f4*>(aco + off);
            const f4 vv = *reinterpret_cast<const f4*>(vis + off);
            gsum4 += av + vv;
        }
        colsum += gsum4;

        // Cross-thread reduction over 64 threads -> hm[b][t].
        const float gsum = (gsum4.x + gsum4.y) + (gsum4.z + gsum4.w);
        red[tid] = gsum;
        __syncthreads();
        #pragma unroll
        for (int st = 32; st > 0; st >>= 1) {
            if (tid < st) red[tid] += red[tid + st];
            __syncthreads();
        }
        if (tid == 0) hm[b * T_DIM + ch * 8 + g] = red[0] * (1.0f / 4096.0f);
        __syncthreads();
    }
    *reinterpret_cast<f4*>(wm_part + (size_t)(b * 8 + ch) * D_DIM + d4) = colsum;
}

// ---------------------------------------------------------------------------
// Pass B0: combine the 8 chunk partials -> wm[b][d] = mean_T of hw.
// ---------------------------------------------------------------------------
__global__ void wm_finalize_kernel(const float* __restrict__ wm_part,
                                   float* __restrict__ wm) {
    const int b  = blockIdx.x;
    const int d4 = threadIdx.x << 2;   // 64 threads * 4 cols
    f4 s = {0.f, 0.f, 0.f, 0.f};
    #pragma unroll
    for (int c = 0; c < 8; ++c)
        s += *reinterpret_cast<const f4*>(wm_part + (size_t)(b * 8 + c) * D_DIM + d4);
    s *= (1.0f / 1024.0f);
    *reinterpret_cast<f4*>(wm + (size_t)b * D_DIM + d4) = s;
}

// ---------------------------------------------------------------------------
// Pass B1/B2: out = sigmoid(A @ W^T + bias) via V_WMMA_F32_16X16X4_F32.
//   A: M x K row-major, W: N x K row-major, out: M x N.
// One wave (32 threads) per 16x16 C tile; K-loop in steps of 4.
// 32-bit operand layout (ISA 7.12.2):
//   A: lane(0..15)=M row, VGPR0 holds K=k0+0 (lanes 0-15) / k0+2 (lanes 16-31),
//      VGPR1 holds k0+1 / k0+3.  B mirrors with lane = N column of W.
//   C: VGPR r -> M = r (lanes 0-15) or r+8 (lanes 16-31); N = lane & 15.
// ---------------------------------------------------------------------------
__global__ void wmma_gemm_bias_sigmoid(const float* __restrict__ A,
                                       const float* __restrict__ W,
                                       const float* __restrict__ bias,
                                       float* __restrict__ out,
                                       int M, int N, int K) {
    const int nTiles = N >> 4;
    const int tile = blockIdx.x;
    const int m0 = (tile / nTiles) << 4;
    const int n0 = (tile % nTiles) << 4;

    const int lane = threadIdx.x;        // 0..31, EXEC all ones
    const int row  = lane & 15;
    const int khi  = (lane >> 4) << 1;   // 0 for lanes 0-15, 2 for lanes 16-31

    const float* __restrict__ arow = A + (m0 + row) * K + khi;
    const float* __restrict__ wrow = W + (n0 + row) * K + khi;

    v8f c = {};
    for (int k0 = 0; k0 < K; k0 += 4) {
        v2f av, bv;
        av.x = arow[k0];
        av.y = arow[k0 + 1];
        bv.x = wrow[k0];
        bv.y = wrow[k0 + 1];
        // 8 args: (neg_a, A, neg_b, B, c_mod, C, reuse_a, reuse_b)
        c = __builtin_amdgcn_wmma_f32_16x16x4_f32(false, av, false, bv,
                                                  (short)0, c, false, false);
    }

    const int n  = n0 + row;
    const float bn = bias[n];
    const int moff = (lane >> 4) << 3;   // +8 for upper half-lanes
    #pragma unroll
    for (int r = 0; r < 8; ++r) {
        const int m = m0 + r + moff;
        out[m * N + n] = sigmoidf(c[r] + bn);
    }
}

// ---------------------------------------------------------------------------
// Pass C: one 64-thread block per (b, t-group). Each thread owns 4 d-columns
// (b128 loads/stores). Re-pools from the tile it must load anyway (reads are
// expected L2 hits after Pass A), forms c_att in registers, applies
// scale = (h + w + c)/3 under the IS_BAG mask. Output stores are NON-TEMPORAL
// so the 128 MB of writes do not evict the still-needed input lines from L2.
// ---------------------------------------------------------------------------
__global__ void apply_scale_kernel(const float* __restrict__ aco,
                                   const float* __restrict__ vis,
                                   const int* __restrict__ bag,
                                   const float* __restrict__ h_att,
                                   const float* __restrict__ w_att,
                                   const float* __restrict__ conv_w,
                                   const float* __restrict__ conv_b,
                                   float* __restrict__ aco_out,
                                   float* __restrict__ vis_out) {
    const int b  = blockIdx.x >> 6;    // 128
    const int t  = blockIdx.x & 63;    // 64 groups
    const int d4 = threadIdx.x << 2;   // 64 threads * 4 cols = 256
    const size_t base = ((size_t)(b * L_DIM + t * G_DIM)) * D_DIM + d4;

    f4 a[8], v[8];
    f4 sa = {0.f, 0.f, 0.f, 0.f};
    f4 sv = {0.f, 0.f, 0.f, 0.f};

    #pragma unroll
    for (int i = 0; i < 8; ++i) {
        a[i] = *reinterpret_cast<const f4*>(aco + base + (size_t)i * D_DIM);
        v[i] = *reinterpret_cast<const f4*>(vis + base + (size_t)i * D_DIM);
        sa += a[i];
        sv += v[i];
    }

    const float w0 = conv_w[0];
    const float w1 = conv_w[1];
    const float cb = conv_b[0];
    const float h  = h_att[b * T_DIM + t];
    const f4 wv = *reinterpret_cast<const f4*>(w_att + (size_t)b * D_DIM + d4);

    f4 sc;
    #pragma unroll
    for (int k = 0; k < 4; ++k) {
        const float ap = sa[k] * 0.125f;          // mean over G=8
        const float vp = sv[k] * 0.125f;
        const float catt = sigmoidf(w0 * ap + w1 * vp + cb);
        sc[k] = (h + wv[k] + catt) * (1.0f / 3.0f);
    }

    #pragma unroll
    for (int i = 0; i < 8; ++i) {
        const int l = t * G_DIM + i;
        const bool msk = (bag[b * L_DIM + l] == 1);
        f4 ao, vo;
        #pragma unroll
        for (int k = 0; k < 4; ++k) {
            ao[k] = msk ? a[i][k] * sc[k] : a[i][k];
            vo[k] = msk ? v[i][k] * sc[k] : v[i][k];
        }
        __builtin_nontemporal_store(
            ao, reinterpret_cast<f4*>(aco_out + base + (size_t)i * D_DIM));
        __builtin_nontemporal_store(
            vo, reinterpret_cast<f4*>(vis_out + base + (size_t)i * D_DIM));
    }
}

// ---------------------------------------------------------------------------
extern "C" void kernel_launch(void* const* d_in, const int* in_sizes, int n_in,
                              void* d_out, int out_size, void* d_ws, size_t ws_size,
                              hipStream_t stream) {
    (void)in_sizes; (void)n_in; (void)out_size; (void)ws_size;

    const float* aco   = (const float*)d_in[0];  // (128,512,256)
    const float* vis   = (const float*)d_in[1];  // (128,512,256)
    const int*   bag   = (const int*)  d_in[2];  // (128,512)
    const float* Wh    = (const float*)d_in[3];  // (64,64)
    const float* bh    = (const float*)d_in[4];  // (64)
    const float* Ww    = (const float*)d_in[5];  // (256,256)
    const float* bw    = (const float*)d_in[6];  // (256)
    const float* convw = (const float*)d_in[7];  // (2)
    const float* convb = (const float*)d_in[8];  // (1)

    // Workspace layout (f32): ~1.3 MB total.
    float* ws      = (float*)d_ws;
    float* hm      = ws;                                   // 128*64
    float* wm_part = hm + B_DIM * T_DIM;                   // 128*8*256
    float* wm      = wm_part + B_DIM * 8 * D_DIM;          // 128*256
    float* h_att   = wm + B_DIM * D_DIM;                   // 128*64
    float* w_att   = h_att + B_DIM * T_DIM;                // 128*256

    float* aco_out = (float*)d_out;
    float* vis_out = aco_out + (size_t)B_DIM * L_DIM * D_DIM;

    // Pass A: fused pooling + reductions (reads both inputs once, warms L2).
    pool_reduce_kernel<<<B_DIM * 8, 64, 0, stream>>>(aco, vis, hm, wm_part);

    // Pass B0: finalize wm.
    wm_finalize_kernel<<<B_DIM, 64, 0, stream>>>(wm_part, wm);

    // Pass B1: h_att = sigmoid(hm @ Wh^T + bh)   (M=128, N=64, K=64)
    wmma_gemm_bias_sigmoid<<<(B_DIM / 16) * (T_DIM / 16), 32, 0, stream>>>(
        hm, Wh, bh, h_att, B_DIM, T_DIM, T_DIM);

    // Pass B2: w_att = sigmoid(wm @ Ww^T + bw)   (M=128, N=256, K=256)
    wmma_gemm_bias_sigmoid<<<(B_DIM / 16) * (D_DIM / 16), 32, 0, stream>>>(
        wm, Ww, bw, w_att, B_DIM, D_DIM, D_DIM);

    // Pass C: fused c_att recompute + scale + masked NT write of both outputs.
    apply_scale_kernel<<<B_DIM * T_DIM, 64, 0, stream>>>(
        aco, vis, bag, h_att, w_att, convw, convb, aco_out, vis_out);
}